// stagenet_references_vis_91250875171450
// MI455X (gfx1250) — compile-verified
//
#include <hip/hip_runtime.h>
#include <cstdint>

// Problem constants from the reference
#define NV 4
#define NB 2
#define NC 32
#define ND 32
#define NH 128
#define NW 160
#define NG 8

// ---------------------------------------------------------------------------
// Setup: per (b, v) compute rot(3x3)+trans(3) of  P = src_proj_new @ inv(ref_proj_new)
// ref/src_proj_new = extr with rows0..2 replaced by K[:3,:3] @ extr[:3,:4]  (row3 = [0,0,0,1])
// Stored in d_ws as 12 floats per (b,v): rot row-major, then trans.
// ---------------------------------------------------------------------------
__global__ void setup_rt_kernel(const float* __restrict__ proj, float* __restrict__ rt)
{
    int t = threadIdx.x;
    if (t >= NB * NV) return;
    int b = t / NV;
    int v = t % NV;

    float Mr[16], Ms[16];
    for (int which = 0; which < 2; ++which) {
        int vv = which == 0 ? 0 : (v + 1);
        float* M = which == 0 ? Mr : Ms;
        const float* E = proj + ((((size_t)b * (NV + 1) + vv) * 2 + 0) * 16); // extrinsics
        const float* K = proj + ((((size_t)b * (NV + 1) + vv) * 2 + 1) * 16); // intrinsics
        for (int r = 0; r < 3; ++r)
            for (int c = 0; c < 4; ++c)
                M[r * 4 + c] = K[r * 4 + 0] * E[0 * 4 + c] +
                               K[r * 4 + 1] * E[1 * 4 + c] +
                               K[r * 4 + 2] * E[2 * 4 + c];
        for (int c = 0; c < 4; ++c) M[12 + c] = E[12 + c]; // row3 = [0,0,0,1]
    }

    // Affine inverse of Mr = [[A, b],[0,1]]:  inv = [[A^-1, -A^-1 b],[0,1]]
    float a00 = Mr[0], a01 = Mr[1], a02 = Mr[2];
    float a10 = Mr[4], a11 = Mr[5], a12 = Mr[6];
    float a20 = Mr[8], a21 = Mr[9], a22 = Mr[10];
    float b0 = Mr[3], b1 = Mr[7], b2 = Mr[11];
    float det = a00 * (a11 * a22 - a12 * a21)
              - a01 * (a10 * a22 - a12 * a20)
              + a02 * (a10 * a21 - a11 * a20);
    float id = 1.0f / det;
    float i00 =  (a11 * a22 - a12 * a21) * id;
    float i01 = -(a01 * a22 - a02 * a21) * id;
    float i02 =  (a01 * a12 - a02 * a11) * id;
    float i10 = -(a10 * a22 - a12 * a20) * id;
    float i11 =  (a00 * a22 - a02 * a20) * id;
    float i12 = -(a00 * a12 - a02 * a10) * id;
    float i20 =  (a10 * a21 - a11 * a20) * id;
    float i21 = -(a00 * a21 - a01 * a20) * id;
    float i22 =  (a00 * a11 - a01 * a10) * id;
    float t0 = -(i00 * b0 + i01 * b1 + i02 * b2);
    float t1 = -(i10 * b0 + i11 * b1 + i12 * b2);
    float t2 = -(i20 * b0 + i21 * b1 + i22 * b2);
    float Mi[16] = { i00, i01, i02, t0,
                     i10, i11, i12, t1,
                     i20, i21, i22, t2,
                     0.f, 0.f, 0.f, 1.f };

    float* o = rt + (size_t)(b * NV + v) * 12;
    for (int r = 0; r < 3; ++r) {
        for (int c = 0; c < 3; ++c) {
            float s = 0.f;
            for (int k = 0; k < 4; ++k) s += Ms[r * 4 + k] * Mi[k * 4 + c];
            o[r * 3 + c] = s;
        }
        float s = 0.f;
        for (int k = 0; k < 4; ++k) s += Ms[r * 4 + k] * Mi[k * 4 + 3];
        o[9 + r] = s;
    }
}

// ---------------------------------------------------------------------------
// Fused main kernel: one wave32 per pixel, lane = depth hypothesis.
// Ref feature vectors double-buffered into LDS via CDNA5 async global->LDS
// copies (ASYNCcnt): view v+1's copy is in flight during view v's gather loop;
// in-order async completion lets "s_wait_asynccnt 1" release the older copy.
// ---------------------------------------------------------------------------
__global__ __launch_bounds__(256) void mvs_fused_kernel(
    const float* __restrict__ ref_feats, const float* __restrict__ src_feats,
    const float* __restrict__ depth_hypo, const float* __restrict__ w_reg,
    const float* __restrict__ rt, float* __restrict__ out)
{
    constexpr int WAVES = 8;
    __shared__ float s_ref[WAVES][2][NC];   // ping-pong per wave

    const int lane = threadIdx.x & 31;
    const int wv   = threadIdx.x >> 5;
    const int pix  = blockIdx.x * WAVES + wv;
    if (pix >= NB * NH * NW) return;
    const int w = pix % NW;
    const int h = (pix / NW) % NH;
    const int b = pix / (NW * NH);
    const int d = lane; // depth hypothesis index

    const float xf = (float)w, yf = (float)h;
    const float dep = depth_hypo[(((size_t)b * ND + d) * NH + h) * NW + w];

    // Raw LDS byte offsets of this lane's two slots (low 32 bits of flat shared addr)
    const uint32_t ldsOff0 = (uint32_t)(uintptr_t)&s_ref[wv][0][lane];
    const uint32_t ldsOff1 = (uint32_t)(uintptr_t)&s_ref[wv][1][lane];

    // Kick off view 0's ref staging immediately.
    {
        const float* gref0 = ref_feats + (((size_t)(0 * NB + b) * NC + lane) * NH + h) * NW + w;
        asm volatile("global_load_async_to_lds_b32 %0, %1, off"
                     :: "v"(ldsOff0), "v"(gref0) : "memory");
    }

    float cw_sum = 1e-8f;
    float cf[NG];
#pragma unroll
    for (int g = 0; g < NG; ++g) cf[g] = 0.f;

#pragma unroll
    for (int v = 0; v < NV; ++v) {
        // Prefetch next view's ref vector into the other slot while this view runs.
        if (v + 1 < NV) {
            const uint32_t ldsN = ((v + 1) & 1) ? ldsOff1 : ldsOff0;
            const float* grefN =
                ref_feats + ((((size_t)(v + 1) * NB + b) * NC + lane) * NH + h) * NW + w;
            asm volatile("global_load_async_to_lds_b32 %0, %1, off"
                         :: "v"(ldsN), "v"(grefN) : "memory");
        }

        const float* R = rt + (size_t)(b * NV + v) * 12;
        float rx = R[0] * xf + R[1] * yf + R[2];
        float ry = R[3] * xf + R[4] * yf + R[5];
        float rz = R[6] * xf + R[7] * yf + R[8];
        float X = rx * dep + R[9];
        float Y = ry * dep + R[10];
        float Z = rz * dep + R[11];
        Z = (Z == 0.0f) ? 1e-9f : Z;
        float invZ = 1.0f / Z;
        float px = X * invZ, py = Y * invZ;
        float x0f = floorf(px), y0f = floorf(py);
        float wx = px - x0f, wy = py - y0f;
        int x0 = (int)x0f, y0 = (int)y0f;
        int x1 = x0 + 1,  y1 = y0 + 1;
        float v00 = (x0 >= 0 && x0 < NW && y0 >= 0 && y0 < NH) ? 1.f : 0.f;
        float v01 = (x1 >= 0 && x1 < NW && y0 >= 0 && y0 < NH) ? 1.f : 0.f;
        float v10 = (x0 >= 0 && x0 < NW && y1 >= 0 && y1 < NH) ? 1.f : 0.f;
        float v11 = (x1 >= 0 && x1 < NW && y1 >= 0 && y1 < NH) ? 1.f : 0.f;
        float w00 = (1.f - wx) * (1.f - wy) * v00;
        float w01 = wx * (1.f - wy) * v01;
        float w10 = (1.f - wx) * wy * v10;
        float w11 = wx * wy * v11;
        int cx0 = min(max(x0, 0), NW - 1), cx1 = min(max(x1, 0), NW - 1);
        int cy0 = min(max(y0, 0), NH - 1), cy1 = min(max(y1, 0), NH - 1);
        int i00 = cy0 * NW + cx0, i01 = cy0 * NW + cx1;
        int i10 = cy1 * NW + cx0, i11 = cy1 * NW + cx1;

        const float* SP = src_feats + (((size_t)v * NB + b) * NC) * (size_t)(NH * NW);

        // Release this view's ref copy. Async loads complete in order, so if the
        // next view's copy is still in flight we only need "<= 1 outstanding".
        if (v + 1 < NV) {
            asm volatile("s_wait_asynccnt 0x1" ::: "memory");
        } else {
            asm volatile("s_wait_asynccnt 0x0" ::: "memory");
        }
        const float* refL = s_ref[wv][v & 1];

        float acc[NG];
#pragma unroll
        for (int g = 0; g < NG; ++g) acc[g] = 0.f;

#pragma unroll 4
        for (int c = 0; c < NC; ++c) {
            const float* p = SP + (size_t)c * (NH * NW);
            // hint next channel plane's tap line into cache (global_prefetch_b8)
            __builtin_prefetch(p + (NH * NW) + i00, 0, 1);
            float f = w00 * p[i00] + w01 * p[i01] + w10 * p[i10] + w11 * p[i11];
            acc[c >> 2] = fmaf(f, refL[c], acc[c >> 2]);
        }

        float corsum = 0.f;
        float cfeat[NG];
#pragma unroll
        for (int g = 0; g < NG; ++g) { cfeat[g] = acc[g] * 0.25f; corsum += cfeat[g]; }

        // cor_weight = max_d softmax_d(corsum) = 1 / sum(exp(x - max))
        float m = corsum;
#pragma unroll
        for (int off = 16; off > 0; off >>= 1) m = fmaxf(m, __shfl_xor(m, off));
        float e = __expf(corsum - m);
        float s = e;
#pragma unroll
        for (int off = 16; off > 0; off >>= 1) s += __shfl_xor(s, off);
        float cw = 1.0f / s;

        cw_sum += cw;
#pragma unroll
        for (int g = 0; g < NG; ++g) cf[g] = fmaf(cw, cfeat[g], cf[g]);
    }

    // Normalize, project onto w_reg, softmax over depth, argmax
    float invs = 1.0f / (cw_sum + 1e-7f);
    float logit = 0.f;
#pragma unroll
    for (int g = 0; g < NG; ++g) logit += (cf[g] * invs) * w_reg[g];

    float m2 = logit;
#pragma unroll
    for (int off = 16; off > 0; off >>= 1) m2 = fmaxf(m2, __shfl_xor(m2, off));
    float e2 = __expf(logit - m2);
    float s2 = e2;
#pragma unroll
    for (int off = 16; off > 0; off >>= 1) s2 += __shfl_xor(s2, off);
    float aw = e2 / s2;

    // attn_weight output: after depth block
    out[(size_t)NB * NH * NW + (((size_t)b * ND + d) * NH + h) * NW + w] = aw;

    // argmax over depth (first max on ties, matching jnp.argmax)
    float bv = aw; int bi = d;
#pragma unroll
    for (int off = 16; off > 0; off >>= 1) {
        float ov = __shfl_xor(bv, off);
        int   oi = __shfl_xor(bi, off);
        if (ov > bv || (ov == bv && oi < bi)) { bv = ov; bi = oi; }
    }
    float dsel = __shfl(dep, bi);
    if (lane == 0) out[((size_t)b * NH + h) * NW + w] = dsel;
}

// ---------------------------------------------------------------------------
extern "C" void kernel_launch(void* const* d_in, const int* in_sizes, int n_in,
                              void* d_out, int out_size, void* d_ws, size_t ws_size,
                              hipStream_t stream)
{
    (void)in_sizes; (void)n_in; (void)out_size; (void)ws_size;
    const float* ref_feats  = (const float*)d_in[0];
    const float* src_feats  = (const float*)d_in[1];
    const float* proj       = (const float*)d_in[2];
    const float* depth_hypo = (const float*)d_in[3];
    const float* w_reg      = (const float*)d_in[4];
    float* out = (float*)d_out;
    float* rt  = (float*)d_ws; // needs NB*NV*12 floats = 384 bytes

    setup_rt_kernel<<<1, 32, 0, stream>>>(proj, rt);

    const int pixels = NB * NH * NW;     // 40960
    const int waves_per_block = 8;       // 256 threads
    mvs_fused_kernel<<<pixels / waves_per_block, 256, 0, stream>>>(
        ref_feats, src_feats, depth_hypo, w_reg, rt, out);
}